// FastfoodTransform_65515431133411
// MI455X (gfx1250) — compile-verified
//
#include <hip/hip_runtime.h>

// Fastfood transform for MI455X (gfx1250, wave32).
// FWHT_1024 = (H16 over i) x (H16 over j) x (H4 over k), n = i*64 + j*4 + k.
// H16 mode-products are done with V_WMMA_F32_16X16X4_F32 (A = H16 slice built
// from popcount signs in registers, B = data fibers from LDS, C accumulates
// over 4 K-chunks). H4 is a 2-stage VALU butterfly. Everything stays in LDS
// between the single global read of x and the single global write of out.

#define IN_EXT  1024
#define NSTACK  4
#define OUTSZ   4096
#define ROWS    8          // batch rows per workgroup
#define BLOCK   256        // 8 waves of 32

typedef __attribute__((ext_vector_type(2))) float v2f;
typedef __attribute__((ext_vector_type(8))) float v8f;

__device__ __forceinline__ float hsign(int r, int c) {
    // Sylvester Hadamard entry H16[r][c] = (-1)^popcount(r&c)
    return (__builtin_popcount(r & c) & 1) ? -1.0f : 1.0f;
}

// One H16 mode-product over a ROWS x 1024 tile living in LDS.
// RS  = row stride (elements) between batch rows in the tile
// STR = element stride along the contracted fiber (4 -> contract j, 64 -> contract i)
template <int RS, int STR>
__device__ __forceinline__ void wmma_pass(float* T) {
    const int lane = threadIdx.x & 31;
    const int wave = threadIdx.x >> 5;
    const int m    = lane & 15;   // A-matrix row / B,D column, per ISA layout
    const int hi   = lane >> 4;   // half-wave select
    const int NTILES = ROWS * 64 / 16;  // 512 (b,fiber) combos -> 32 tiles

    for (int tile = wave; tile < NTILES; tile += BLOCK / 32) {
        int combo = tile * 16 + m;          // combo = b*64 + (fiber coords)
        int b   = combo >> 6;
        int rem = combo & 63;
        int base;
        if (STR == 4) {
            // contract over j: rem = i*4 + k ; addr = b*RS + i*64 + k + j*4
            base = b * RS + (rem >> 2) * 64 + (rem & 3);
        } else {
            // contract over i: rem = j*4 + k ; addr = b*RS + j*4 + k + i*64
            base = b * RS + rem;
        }

        v8f acc = {};
#pragma unroll
        for (int ch = 0; ch < 4; ++ch) {
            int c0 = ch * 4 + hi * 2;       // absolute K columns for v0/v1
            v2f a, d;
            a.x = hsign(m, c0);
            a.y = hsign(m, c0 + 1);
            d.x = T[base + c0 * STR];
            d.y = T[base + (c0 + 1) * STR];
            acc = __builtin_amdgcn_wmma_f32_16x16x4_f32(
                false, a, false, d, (short)0, acc, false, false);
        }
#pragma unroll
        for (int r = 0; r < 8; ++r) {
            T[base + (r + 8 * hi) * STR] = acc[r];  // D row M = r + 8*hi
        }
    }
}

// Full FWHT_1024 on ROWS rows (row stride RS) in LDS; caller synced before.
template <int RS>
__device__ __forceinline__ void fwht_rows(float* T) {
    // H4 over k (2-stage butterfly on 4 consecutive elements)
    for (int f = threadIdx.x; f < ROWS * 256; f += BLOCK) {
        int b = f >> 8, w = f & 255;       // w = i*16 + j
        float* p = T + b * RS + w * 4;
        float t0 = p[0], t1 = p[1], t2 = p[2], t3 = p[3];
        float s0 = t0 + t1, d0 = t0 - t1, s1 = t2 + t3, d1 = t2 - t3;
        p[0] = s0 + s1; p[1] = d0 + d1; p[2] = s0 - s1; p[3] = d0 - d1;
    }
    __syncthreads();
    wmma_pass<RS, 4>(T);    // H16 over j
    __syncthreads();
    wmma_pass<RS, 64>(T);   // H16 over i
    __syncthreads();
}

__global__ void __launch_bounds__(BLOCK)
fastfood_kernel(const float* __restrict__ x,
                const float* __restrict__ Bm,
                const float* __restrict__ Gm,
                const float* __restrict__ Sm,
                const float* __restrict__ bias,
                const int*   __restrict__ P,
                float* __restrict__ out,
                int batch)
{
    __shared__ float xbuf[ROWS * IN_EXT];   //  32 KB: x tile
    __shared__ float buf1[ROWS * OUTSZ];    // 128 KB: stage-1 FWHT results, all stacks
    __shared__ float wbuf[ROWS * IN_EXT];   //  32 KB: stage-2 working buffer

    const int row0 = blockIdx.x * ROWS;

    // ---- load x tile (coalesced float4) ----
    {
        const float4* xs = (const float4*)(x + (size_t)row0 * IN_EXT);
        float4* xd = (float4*)xbuf;
        const int n4 = ROWS * IN_EXT / 4;   // 2048
        for (int idx = threadIdx.x; idx < n4; idx += BLOCK) {
            int r = idx >> 8;               // 256 float4 per row
            float4 v = make_float4(0.f, 0.f, 0.f, 0.f);
            if (row0 + r < batch) v = xs[idx];
            xd[idx] = v;
        }
    }
    __syncthreads();

    // ---- stage 1: per stack, y = B[s]*x, FWHT, keep all stacks in buf1 ----
    for (int s = 0; s < NSTACK; ++s) {
        for (int idx = threadIdx.x; idx < ROWS * IN_EXT; idx += BLOCK) {
            int b = idx >> 10, n = idx & 1023;
            buf1[b * OUTSZ + s * IN_EXT + n] = Bm[s * IN_EXT + n] * xbuf[idx];
        }
        __syncthreads();
        fwht_rows<OUTSZ>(buf1 + s * IN_EXT);   // ends with syncthreads
    }

    // ---- stage 2: per stack, gather(P)*G, FWHT, write S*y + bias ----
    for (int s = 0; s < NSTACK; ++s) {
        for (int idx = threadIdx.x; idx < ROWS * IN_EXT; idx += BLOCK) {
            int b = idx >> 10, n = idx & 1023;
            int o = s * IN_EXT + n;
            wbuf[idx] = Gm[o] * buf1[b * OUTSZ + P[o]];
        }
        __syncthreads();
        fwht_rows<IN_EXT>(wbuf);               // ends with syncthreads
        for (int idx = threadIdx.x; idx < ROWS * IN_EXT; idx += BLOCK) {
            int b = idx >> 10, n = idx & 1023;
            int o = s * IN_EXT + n;
            int gr = row0 + b;
            if (gr < batch)
                out[(size_t)gr * OUTSZ + o] = Sm[o] * wbuf[idx] + bias[o];
        }
        __syncthreads();   // WAR: wbuf reused next stack
    }
}

extern "C" void kernel_launch(void* const* d_in, const int* in_sizes, int n_in,
                              void* d_out, int out_size, void* d_ws, size_t ws_size,
                              hipStream_t stream) {
    const float* x    = (const float*)d_in[0];
    const float* Bm   = (const float*)d_in[1];
    const float* Gm   = (const float*)d_in[2];
    const float* Sm   = (const float*)d_in[3];
    const float* bias = (const float*)d_in[4];
    const int*   P    = (const int*)d_in[5];
    float* out = (float*)d_out;

    int batch = in_sizes[0] / IN_EXT;
    int grid  = (batch + ROWS - 1) / ROWS;
    fastfood_kernel<<<dim3(grid), dim3(BLOCK), 0, stream>>>(
        x, Bm, Gm, Sm, bias, P, out, batch);
}